// SAGE_35802847379700
// MI455X (gfx1250) — compile-verified
//
#include <hip/hip_runtime.h>

#define H 128
#define PADN 144  // padded N-stride (in b64 pairs) for conflict-free ds_load_b64

typedef float v2f __attribute__((ext_vector_type(2)));
typedef float v8f __attribute__((ext_vector_type(8)));

// ---------------------------------------------------------------- utilities
__global__ void fill_i32(int* __restrict__ p, int v, long n) {
  long i = (long)blockIdx.x * blockDim.x + threadIdx.x;
  long stride = (long)gridDim.x * blockDim.x;
  for (; i < n; i += stride) p[i] = v;
}

__global__ void deg_count(const int* __restrict__ dst, int* __restrict__ deg, int E) {
  int e = blockIdx.x * blockDim.x + threadIdx.x;
  if (e < E) atomicAdd(&deg[dst[e]], 1);
}

__global__ void inv_from_deg(const int* __restrict__ deg, float* __restrict__ invdeg, int N) {
  int i = blockIdx.x * blockDim.x + threadIdx.x;
  if (i < N) invdeg[i] = 1.0f / fmaxf((float)deg[i], 1.0f);
}

// ------------------------------------------------- deterministic excl. scan
// level 1: per-block exclusive scan of 256 elements + block sum
__global__ void scan1(const int* __restrict__ deg, int* __restrict__ out,
                      int* __restrict__ bsums, int N) {
  __shared__ int tmp[256];
  int t = threadIdx.x;
  int i = blockIdx.x * 256 + t;
  int v = (i < N) ? deg[i] : 0;
  tmp[t] = v;
  __syncthreads();
  for (int off = 1; off < 256; off <<= 1) {
    int add = (t >= off) ? tmp[t - off] : 0;
    __syncthreads();
    tmp[t] += add;
    __syncthreads();
  }
  if (i < N) out[i] = tmp[t] - v;  // exclusive
  if (t == 255) bsums[blockIdx.x] = tmp[255];
}

// level 2: single-block exclusive scan of block sums (nb <= 1024)
__global__ void scan2(int* __restrict__ bsums, int nb) {
  __shared__ int tmp[1024];
  int t = threadIdx.x;
  int v = (t < nb) ? bsums[t] : 0;
  tmp[t] = v;
  __syncthreads();
  for (int off = 1; off < 1024; off <<= 1) {
    int add = (t >= off) ? tmp[t - off] : 0;
    __syncthreads();
    tmp[t] += add;
    __syncthreads();
  }
  if (t < nb) bsums[t] = tmp[t] - v;
}

// level 3: add block offsets
__global__ void scan3(int* __restrict__ out, const int* __restrict__ bsums, int N) {
  int i = blockIdx.x * 256 + threadIdx.x;
  if (i < N) out[i] += bsums[blockIdx.x];
}

// bucket edges by destination: csr_src[row_start[d] + pos] = src
__global__ void csr_fill(const int* __restrict__ src, const int* __restrict__ dst,
                         const int* __restrict__ row_start, int* __restrict__ cursor,
                         int* __restrict__ csr_src, int E) {
  int e = blockIdx.x * blockDim.x + threadIdx.x;
  if (e < E) {
    int d = dst[e];
    int pos = atomicAdd(&cursor[d], 1);
    csr_src[row_start[d] + pos] = src[e];
  }
}

// x[i] = emb[node_id[i]]   (32 threads per node, float4 per thread)
__global__ void gather_x(const int* __restrict__ node_id, const float* __restrict__ emb,
                         float* __restrict__ x, int N) {
  int t = blockIdx.x * blockDim.x + threadIdx.x;
  int i = t >> 5;
  int c = (t & 31) << 2;
  if (i < N) {
    float4 v = *(const float4*)(emb + (long)node_id[i] * H + c);
    *(float4*)(x + (long)i * H + c) = v;
  }
}

// aggr[i] = sum over neighbors of h[src]  (one wave per node, register acc,
// single write per output element; no fp atomics)
__global__ void csr_aggregate(const float* __restrict__ h, const int* __restrict__ csr_src,
                              const int* __restrict__ row_start, const int* __restrict__ deg,
                              float* __restrict__ aggr, int N) {
  long t = (long)blockIdx.x * blockDim.x + threadIdx.x;
  int i = (int)(t >> 5);
  int c = ((int)t & 31) << 2;
  if (i >= N) return;
  int beg = row_start[i];
  int end = beg + deg[i];
  float4 acc = {0.f, 0.f, 0.f, 0.f};
  for (int j = beg; j < end; ++j) {
    int s = csr_src[j];  // uniform across wave -> broadcast
    float4 v = *(const float4*)(h + (long)s * H + c);
    acc.x += v.x;
    acc.y += v.y;
    acc.z += v.z;
    acc.w += v.w;
  }
  *(float4*)(aggr + (long)i * H + c) = acc;
}

// ---------------------------------------------------------------- WMMA GEMM
// Stage W (row-major [n][k]) into LDS as B-fragment pairs:
//   P[p][n] = { W[n][2p], W[n][2p+1] },  p = 0..63, padded row stride PADN
// so a lane's B fragment (K=ko, K=ko+1 at column n) is a single ds_load_b64.
__device__ __forceinline__ void stage_weights(const float* __restrict__ W,
                                              float* __restrict__ smem, int t) {
  for (int i = t; i < H * (H / 2); i += 256) {
    int n = i >> 6;  // 0..127 : weight row (output column)
    int p = i & 63;  // 0..63  : K pair
    v2f w = *(const v2f*)(W + n * H + 2 * p);  // coalesced global b64
    *(v2f*)&smem[(p * PADN + n) * 2] = w;      // ds_store_b64
  }
}

// One pass of acc[nt] += A x B over full K.  A fragment loaded once per
// k-step and reused across all 8 N-tiles; B fragments from LDS.
template <bool SCALE>
__device__ __forceinline__ void gemm_pass(const float* __restrict__ arow, float scale,
                                          const float* __restrict__ smem, int half,
                                          int l16, v8f acc[8]) {
  for (int kk = 0; kk < H; kk += 4) {
    const int ko = kk + 2 * half;      // this lane's K pair start (A & B frag)
    v2f a = *(const v2f*)(arow + ko);  // A: M=l16 rows, K=ko..ko+1
    if (SCALE) a *= scale;             // fold 1/deg into A fragment
    const int p2 = (kk >> 1) + half;   // pair index in LDS
    const float* base = smem + (p2 * PADN + l16) * 2;
#pragma unroll
    for (int nt = 0; nt < 8; ++nt) {
      v2f b = *(const v2f*)(base + nt * 32);  // ds_load_b64: B[K=ko..+1][N=nt*16+l16]
      acc[nt] = __builtin_amdgcn_wmma_f32_16x16x4_f32(false, a, false, b,
                                                      (short)0, acc[nt], false, false);
    }
  }
}

// out[m][n] = maybe_relu( (aggr[m]*inv_deg[m]) @ Wl^T + bl + h[m] @ Wr^T )
// 256 threads = 8 waves; wave w owns rows m_base..m_base+15, all 128 columns.
__global__ __launch_bounds__(256) void sage_gemm_wmma(
    const float* __restrict__ h, const float* __restrict__ aggr,
    const float* __restrict__ inv_deg, const float* __restrict__ Wl,
    const float* __restrict__ bl, const float* __restrict__ Wr,
    float* __restrict__ out, int N, int do_relu) {
  extern __shared__ float smem[];  // 64 * PADN * 2 floats = 72 KB

  const int t    = threadIdx.x;
  const int lane = t & 31;
  const int wave = t >> 5;
  const int l16  = lane & 15;
  const int half = lane >> 4;

  const int m_base = blockIdx.x * 128 + wave * 16;
  const int row    = m_base + l16;
  const int rowc   = row < N ? row : (N - 1);  // clamp loads; stores predicated

  const float scale = inv_deg[rowc];
  const float* __restrict__ aggr_row = aggr + (long)rowc * H;
  const float* __restrict__ h_row    = h    + (long)rowc * H;

  // warm L0/L2 for this wave's A strips (global_prefetch_b8)
  __builtin_prefetch(aggr_row, 0, 3);
  __builtin_prefetch(h_row, 0, 3);

  v8f acc[8];
#pragma unroll
  for (int nt = 0; nt < 8; ++nt) acc[nt] = (v8f){};

  // pass 1: mean-aggregated neighbors x Wl
  stage_weights(Wl, smem, t);
  __syncthreads();
  gemm_pass<true>(aggr_row, scale, smem, half, l16, acc);
  __syncthreads();

  // pass 2: self features x Wr (reuse LDS)
  stage_weights(Wr, smem, t);
  __syncthreads();
  gemm_pass<false>(h_row, 1.0f, smem, half, l16, acc);

  // epilogue: bias + relu + predicated store
#pragma unroll
  for (int nt = 0; nt < 8; ++nt) {
    const int ncol = nt * 16 + l16;
    const float bias = bl[ncol];
#pragma unroll
    for (int r = 0; r < 8; ++r) {
      int m = m_base + r + 8 * half;  // C/D layout: lane half selects M+8
      float v = acc[nt][r] + bias;
      if (do_relu) v = fmaxf(v, 0.0f);
      if (m < N) out[(long)m * H + ncol] = v;
    }
  }
}

// ---------------------------------------------------------------- launcher
extern "C" void kernel_launch(void* const* d_in, const int* in_sizes, int n_in,
                              void* d_out, int out_size, void* d_ws, size_t ws_size,
                              hipStream_t stream) {
  const int*   node_id = (const int*)d_in[0];
  const int*   edge    = (const int*)d_in[1];
  const float* emb     = (const float*)d_in[2];
  const float* W1l = (const float*)d_in[3];
  const float* b1l = (const float*)d_in[4];
  const float* W1r = (const float*)d_in[5];
  const float* W2l = (const float*)d_in[6];
  const float* b2l = (const float*)d_in[7];
  const float* W2r = (const float*)d_in[8];
  const float* W3l = (const float*)d_in[9];
  const float* b3l = (const float*)d_in[10];
  const float* W3r = (const float*)d_in[11];

  const int N = in_sizes[0];
  const int E = in_sizes[1] / 2;
  const int* src = edge;
  const int* dst = edge + E;
  float* out = (float*)d_out;

  const long NH = (long)N * H;
  const int nb1 = (N + 255) / 256;  // scan blocks (must be <= 1024; 391 here)

  // workspace layout (4-byte elements)
  char* ws = (char*)d_ws;
  float* invdeg    = (float*)ws;                    // [N]
  int*   deg       = (int*)(invdeg + N);            // [N]
  int*   row_start = deg + N;                       // [N]
  int*   cursor    = row_start + N;                 // [N]
  int*   bsums     = cursor + N;                    // [1024]
  int*   csr_src   = bsums + 1024;                  // [E]
  float* aggr      = (float*)(csr_src + E);         // [N*H]
  float* hX        = aggr + NH;                     // [N*H]
  float* hA        = hX + NH;                       // [N*H]

  const int T = 256;
  const int gemm_blocks = (N + 127) / 128;
  const size_t lds_bytes = (size_t)64 * PADN * 2 * sizeof(float);  // 73728

  // ---- CSR build (once; reused by all 3 layers) ----
  fill_i32<<<256, T, 0, stream>>>(deg, 0, N);
  fill_i32<<<256, T, 0, stream>>>(cursor, 0, N);
  deg_count<<<(E + T - 1) / T, T, 0, stream>>>(dst, deg, E);
  inv_from_deg<<<(N + T - 1) / T, T, 0, stream>>>(deg, invdeg, N);
  scan1<<<nb1, 256, 0, stream>>>(deg, row_start, bsums, N);
  scan2<<<1, 1024, 0, stream>>>(bsums, nb1);
  scan3<<<nb1, 256, 0, stream>>>(row_start, bsums, N);
  csr_fill<<<(E + T - 1) / T, T, 0, stream>>>(src, dst, row_start, cursor, csr_src, E);

  // ---- x = emb[node_id] ----
  {
    long threads = (long)N * 32;
    gather_x<<<(int)((threads + T - 1) / T), T, 0, stream>>>(node_id, emb, hX, N);
  }

  const long agg_threads = (long)N * 32;
  const int agg_blocks = (int)((agg_threads + T - 1) / T);

  // ---- layer 1: hX -> hA (relu) ----
  csr_aggregate<<<agg_blocks, T, 0, stream>>>(hX, csr_src, row_start, deg, aggr, N);
  sage_gemm_wmma<<<gemm_blocks, T, lds_bytes, stream>>>(hX, aggr, invdeg, W1l, b1l, W1r,
                                                        hA, N, 1);

  // ---- layer 2: hA -> hX (relu) ----
  csr_aggregate<<<agg_blocks, T, 0, stream>>>(hA, csr_src, row_start, deg, aggr, N);
  sage_gemm_wmma<<<gemm_blocks, T, lds_bytes, stream>>>(hA, aggr, invdeg, W2l, b2l, W2r,
                                                        hX, N, 1);

  // ---- layer 3: hX -> out (no relu) ----
  csr_aggregate<<<agg_blocks, T, 0, stream>>>(hX, csr_src, row_start, deg, aggr, N);
  sage_gemm_wmma<<<gemm_blocks, T, lds_bytes, stream>>>(hX, aggr, invdeg, W3l, b3l, W3r,
                                                        out, N, 0);
}